// AttentionMemoryCell_764504178980
// MI455X (gfx1250) — compile-verified
//
#include <hip/hip_runtime.h>
#include <math.h>

// Problem constants (fixed by setup_inputs): mem (2048,64,512) f32, h_t (64,512),
// w_t (1024,1). Output m_t (64,512) f32.
//
// NOTE: h_t (d_in[0]) and w_t[0:512] cancel exactly out of the sparse-attention
// math (a per-column constant shift is removed by the top-k threshold), so they
// are intentionally unused.

#define M_SZ 2048
#define B_SZ 64
#define H_SZ 512
#define EPSV 1e-7f

typedef __attribute__((ext_vector_type(2))) float v2f;
typedef __attribute__((ext_vector_type(8))) float v8f;
typedef __attribute__((address_space(3))) float lds_f;

#if defined(__has_builtin)
#if __has_builtin(__builtin_amdgcn_tanhf)
#define FAST_TANH(x) __builtin_amdgcn_tanhf(x)
#elif __has_builtin(__builtin_amdgcn_tanh_f32)
#define FAST_TANH(x) __builtin_amdgcn_tanh_f32(x)
#endif
#endif
#ifndef FAST_TANH
#define FAST_TANH(x) tanhf(x)
#endif

// ---------------------------------------------------------------------------
// Kernel 1: score[b*M + m] = sum_d tanh(mem[m,b,d]) * w_t[H + d]
// Each wave32 owns 4 consecutive (b,m) rows (2 KB each, contiguous) and
// double-buffers them through LDS with GLOBAL_LOAD_ASYNC_TO_LDS_B128
// (ASYNCcnt-tracked; in-order completion => s_wait_asynccnt 4 == "previous
// row landed"). Consumption is ds_load_b128 + hw tanh + FMA + wave reduce.
// 256 MB single stream @ 23.3 TB/s ~= 11 us; tanh VALU hides under it.
// ---------------------------------------------------------------------------
#define ROWS_PER_WAVE 4

__device__ __forceinline__ void async_row_to_lds(const float* src, lds_f* dst,
                                                 int lane) {
#pragma unroll
  for (int i = 0; i < 4; ++i) {  // 4 x (32 lanes x 16B) = 2 KB row
    unsigned loff = (unsigned)(uintptr_t)(dst + i * 128 + lane * 4);
    const float* gptr = src + i * 128 + lane * 4;
    asm volatile("global_load_async_to_lds_b128 %0, %1, off"
                 :
                 : "v"(loff), "v"(gptr)
                 : "memory");
  }
}

__global__ __launch_bounds__(256) void score_kernel(const float* __restrict__ mem,
                                                    const float* __restrict__ w_t,
                                                    float* __restrict__ score) {
  __shared__ float stage[8][2][H_SZ];  // 8 waves x double buffer x 2KB = 32 KB
  const int wave = threadIdx.x >> 5;
  const int lane = threadIdx.x & 31;
  const int waveId = blockIdx.x * 8 + wave;
  const int row0 = waveId * ROWS_PER_WAVE;  // linear row index = b*M + m
  const int b = row0 >> 11;                 // M_SZ == 2048 == 1<<11
  const int m0 = row0 & (M_SZ - 1);         // 4 rows never cross a b boundary
  const float* __restrict__ w2 = w_t + H_SZ;

  // AS3 pointers only for the async-DMA destination operand (32-bit LDS addr).
  lds_f* buf[2];
  buf[0] = (lds_f*)&stage[wave][0][0];
  buf[1] = (lds_f*)&stage[wave][1][0];

  // Preload this lane's slice of w2 (2 KB, L2-resident across all waves).
  float4 wv[4];
#pragma unroll
  for (int i = 0; i < 4; ++i)
    wv[i] = *reinterpret_cast<const float4*>(w2 + i * 128 + lane * 4);

  // Prologue: kick off row 0.
  async_row_to_lds(mem + ((size_t)m0 * B_SZ + b) * H_SZ, buf[0], lane);

#pragma unroll
  for (int j = 0; j < ROWS_PER_WAVE; ++j) {
    if (j + 1 < ROWS_PER_WAVE) {  // issue next row into the other buffer
      async_row_to_lds(mem + ((size_t)(m0 + j + 1) * B_SZ + b) * H_SZ,
                       buf[(j + 1) & 1], lane);
      asm volatile("s_wait_asynccnt 0x4" ::: "memory");  // row j complete
    } else {
      asm volatile("s_wait_asynccnt 0x0" ::: "memory");
    }
    __builtin_amdgcn_wave_barrier();

    // Consume through the generic __shared__ pointer (lowers to ds_load_b128);
    // keeps the host pass away from address-space-qualified HIP_vector_type.
    const float* src = &stage[wave][j & 1][0];
    float acc = 0.f;
#pragma unroll
    for (int i = 0; i < 4; ++i) {
      const float4 v = *reinterpret_cast<const float4*>(src + i * 128 + lane * 4);
      acc += FAST_TANH(v.x) * wv[i].x;
      acc += FAST_TANH(v.y) * wv[i].y;
      acc += FAST_TANH(v.z) * wv[i].z;
      acc += FAST_TANH(v.w) * wv[i].w;
    }
#pragma unroll
    for (int off = 16; off > 0; off >>= 1) acc += __shfl_xor(acc, off, 32);
    if (lane == 0) score[(size_t)b * M_SZ + m0 + j] = acc;
  }
}

// ---------------------------------------------------------------------------
// Kernel 2: per column b, delta = 5th-largest(score[:,b]) + eps (argmax-by-index
// x5 so duplicate values count with multiplicity, matching lax.top_k), then
// wnorm = clip(score - delta, 0) / (sum + eps). Tiny: 512 KB read + 512 KB written.
// ---------------------------------------------------------------------------
__global__ __launch_bounds__(256) void topk_norm_kernel(const float* __restrict__ score,
                                                        float* __restrict__ wnorm) {
  __shared__ float arr[M_SZ];
  __shared__ float red[256];
  __shared__ int   redi[256];
  __shared__ float s_delta, s_inv;
  const int b = blockIdx.x;
  const int t = threadIdx.x;
  const float* __restrict__ col = score + (size_t)b * M_SZ;
#pragma unroll
  for (int i = 0; i < M_SZ / 256; ++i) arr[t + 256 * i] = col[t + 256 * i];
  __syncthreads();
  for (int k = 0; k < 5; ++k) {
    float best = -__builtin_inff();
    int bi = 0;
#pragma unroll
    for (int i = 0; i < M_SZ / 256; ++i) {
      const float v = arr[t + 256 * i];
      if (v > best) { best = v; bi = t + 256 * i; }
    }
    red[t] = best; redi[t] = bi;
    __syncthreads();
    for (int s = 128; s > 0; s >>= 1) {
      if (t < s && red[t + s] > red[t]) { red[t] = red[t + s]; redi[t] = redi[t + s]; }
      __syncthreads();
    }
    if (t == 0) {
      if (k == 4) s_delta = red[0] + EPSV;  // 5th largest
      arr[redi[0]] = -__builtin_inff();     // knock out by index
    }
    __syncthreads();
  }
  const float delta = s_delta;
  float lsum = 0.f;
#pragma unroll
  for (int i = 0; i < M_SZ / 256; ++i) lsum += fmaxf(col[t + 256 * i] - delta, 0.f);
  red[t] = lsum;
  __syncthreads();
  for (int s = 128; s > 0; s >>= 1) {
    if (t < s) red[t] += red[t + s];
    __syncthreads();
  }
  if (t == 0) s_inv = 1.f / (red[0] + EPSV);
  __syncthreads();
  const float inv = s_inv;
#pragma unroll
  for (int i = 0; i < M_SZ / 256; ++i) {
    const int m = t + 256 * i;
    wnorm[(size_t)b * M_SZ + m] = fmaxf(col[m] - delta, 0.f) * inv;
  }
}

// ---------------------------------------------------------------------------
// Kernel 3: m_t[b,h] = sum_m wnorm[b,m] * mem[m,b,h]  via V_WMMA_F32_16X16X4_F32.
// A-fragment = the 4 chunk weights broadcast to all 16 rows (all D rows equal,
// row 0 is the result); B-fragment = mem[m0:m0+4, b, h0:h0+16]. Accumulates in
// the f32 C operand over the full m loop (exact f32 math, just a different
// summation order). Two waves/block cover adjacent h-tiles so each 128B line
// is consumed within one block. Second 256 MB stream of mem -> HBM-bound.
// A layout (32-bit A 16x4): lanes 0-15 hold K=0(v0),K=1(v1); lanes 16-31 K=2,K=3.
// ---------------------------------------------------------------------------
__global__ __launch_bounds__(64) void wsum_wmma_kernel(const float* __restrict__ mem,
                                                       const float* __restrict__ wnorm,
                                                       float* __restrict__ out) {
  const int b    = blockIdx.y;
  const int wave = threadIdx.x >> 5;
  const int lane = threadIdx.x & 31;
  const int h0   = blockIdx.x * 32 + wave * 16;
  const int n    = lane & 15;
  const int hi   = lane >> 4;
  const float* __restrict__ wb = wnorm + (size_t)b * M_SZ;
  const size_t colbase   = (size_t)b * H_SZ + h0 + n;
  const size_t rowstride = (size_t)B_SZ * H_SZ;
  v8f c = {0.f, 0.f, 0.f, 0.f, 0.f, 0.f, 0.f, 0.f};
#pragma unroll 4
  for (int m0 = 0; m0 < M_SZ; m0 += 4) {
    const int ka = m0 + 2 * hi;  // this lane-half's K rows: {ka, ka+1}
    v2f a;
    a.x = wb[ka];
    a.y = wb[ka + 1];
    v2f bf;
    bf.x = mem[(size_t)ka * rowstride + colbase];
    bf.y = mem[(size_t)(ka + 1) * rowstride + colbase];
    __builtin_prefetch(mem + (size_t)(ka + 8) * rowstride + colbase, 0, 0);
    // (neg_a, A, neg_b, B, c_mod, C, reuse_a, reuse_b)
    c = __builtin_amdgcn_wmma_f32_16x16x4_f32(false, a, false, bf, (short)0, c,
                                              false, false);
  }
  // D rows are all identical; row 0 lives in VGPR0 of lanes 0-15.
  if (hi == 0) out[(size_t)b * H_SZ + h0 + n] = c[0];
}

// ---------------------------------------------------------------------------
extern "C" void kernel_launch(void* const* d_in, const int* in_sizes, int n_in,
                              void* d_out, int out_size, void* d_ws, size_t ws_size,
                              hipStream_t stream) {
  (void)in_sizes; (void)n_in; (void)out_size; (void)ws_size;
  // d_in[0] = h_t  : unused (cancels exactly, see header comment)
  const float* mem = (const float*)d_in[1];
  const float* w_t = (const float*)d_in[2];
  float* out = (float*)d_out;

  float* score = (float*)d_ws;                 // M*B floats (512 KB)
  float* wnorm = score + (size_t)M_SZ * B_SZ;  // M*B floats (512 KB)

  // 8 waves/block, 4 rows per wave.
  score_kernel<<<(M_SZ * B_SZ) / (8 * ROWS_PER_WAVE), 256, 0, stream>>>(mem, w_t,
                                                                        score);
  // one block per batch column
  topk_norm_kernel<<<B_SZ, 256, 0, stream>>>(score, wnorm);
  // (h-tiles of 32, b) blocks of 2 waves
  wsum_wmma_kernel<<<dim3(H_SZ / 32, B_SZ), 64, 0, stream>>>(mem, wnorm, out);
}